// VertexConv_39144331936316
// MI455X (gfx1250) — compile-verified
//
#include <hip/hip_runtime.h>

// Problem constants (from reference): N=32, C=64, T=128, V=25, K=10
#define N_ 32
#define C_ 64
#define T_ 128
#define V_ 25
#define K_ 10
#define M_ (N_ * V_)   // 800 rows, divisible by 16

typedef float v2f __attribute__((ext_vector_type(2)));
typedef float v8f __attribute__((ext_vector_type(8)));

// ---------------------------------------------------------------------------
// Stage 1: mf2[k, m=n*V+v, c] = mean over T of region[n,c,t,v,k]
// One block per (n,c); thread idx = v*K+k; coalesced b32 streaming loads
// (each wave-load covers exactly one 128B line).
// ---------------------------------------------------------------------------
__global__ void vc_mean_kernel(const float* __restrict__ region,
                               float* __restrict__ mf2) {
    const int nc = blockIdx.x;          // n*C + c
    const int n  = nc / C_;
    const int c  = nc % C_;
    const int idx = threadIdx.x;        // v*K + k
    if (idx >= V_ * K_) return;

    const float* base = region + (size_t)nc * T_ * V_ * K_ + idx;
    float s = 0.0f;
    #pragma unroll 4
    for (int t = 0; t < T_; ++t)
        s += base[(size_t)t * (V_ * K_)];

    const int v = idx / K_;
    const int k = idx % K_;
    const int m = n * V_ + v;
    mf2[((size_t)k * M_ + m) * C_ + c] = s * (1.0f / (float)T_);
}

// ---------------------------------------------------------------------------
// Stage 2: per-k GEMM logits = mf2_k(800x64) * w1[k]^T(64x10->pad16) via
// V_WMMA_F32_16X16X4_F32 (exact fp32), then +b1, softmax over p,
// wk2[k, m] = sum_p softmax_p * w2[p].
// One wave (32 threads) per 16-row tile; 50 tiles per k, 500 blocks total.
// ---------------------------------------------------------------------------
__global__ void vc_attn_kernel(const float* __restrict__ mf2,
                               const float* __restrict__ w1,   // (K, K, C)
                               const float* __restrict__ b1,   // (K, K)
                               const float* __restrict__ w2,   // (K)
                               float* __restrict__ wk2) {      // (K, M)
    const int tile = blockIdx.x;              // 0..499
    const int k    = tile / (M_ / 16);        // 0..9
    const int m0   = (tile % (M_ / 16)) * 16; // row base
    const int lane = threadIdx.x;             // 0..31
    const int half = lane >> 4;               // 0: lanes 0-15, 1: lanes 16-31
    const int l    = lane & 15;

    // A-row pointer for this lane's matrix row (row = l in both halves)
    const float* arow = mf2 + ((size_t)k * M_ + (m0 + l)) * C_;
    // B column p = l (valid p < 10). Clamp the row index so every lane loads
    // from valid memory (no divergent branches), then zero lanes p >= 10 with
    // a mask multiply -> clean global_load_b64 + v_cndmask, no exec juggling.
    const int   pclamp = (l < K_) ? l : (K_ - 1);
    const float bmask  = (l < K_) ? 1.0f : 0.0f;
    const float* brow  = w1 + ((size_t)k * K_ + pclamp) * C_;

    v8f acc = {0.f, 0.f, 0.f, 0.f, 0.f, 0.f, 0.f, 0.f};

    // 16 WMMA steps over C (4 per step).
    // A frag: lanes 0-15 hold {A[l,c0],A[l,c0+1]}, lanes 16-31 {A[l,c0+2],A[l,c0+3]}
    // B frag: lanes 0-15 hold {B[c0,l],B[c0+1,l]}, lanes 16-31 {B[c0+2,l],B[c0+3,l]}
    #pragma unroll
    for (int j = 0; j < C_ / 4; ++j) {
        const int c0 = j * 4 + half * 2;
        v2f a, b;
        a.x = arow[c0];
        a.y = arow[c0 + 1];
        b.x = brow[c0] * bmask;
        b.y = brow[c0 + 1] * bmask;
        acc = __builtin_amdgcn_wmma_f32_16x16x4_f32(
            /*neg_a=*/false, a, /*neg_b=*/false, b,
            /*c_mod=*/(short)0, acc, /*reuse_a=*/false, /*reuse_b=*/false);
    }

    // D layout: lane holds column (l), rows g (+8 for upper half) in acc[g].
    __shared__ float tile_lds[16][17];  // +1 pad against bank conflicts
    #pragma unroll
    for (int g = 0; g < 8; ++g)
        tile_lds[g + half * 8][l] = acc[g];
    __syncthreads();

    // Lanes 0-15: one row each -> bias, softmax over p, fold w2.
    if (lane < 16) {
        const int row = lane;
        float lg[K_];
        float mx = -1e30f;
        #pragma unroll
        for (int p = 0; p < K_; ++p) {
            lg[p] = tile_lds[row][p] + b1[k * K_ + p];
            mx = fmaxf(mx, lg[p]);
        }
        float sum = 0.0f;
        #pragma unroll
        for (int p = 0; p < K_; ++p) {
            lg[p] = __expf(lg[p] - mx);
            sum += lg[p];
        }
        const float inv = 1.0f / sum;
        float wkv = 0.0f;
        #pragma unroll
        for (int p = 0; p < K_; ++p)
            wkv += lg[p] * inv * w2[p];
        wk2[(size_t)k * M_ + m0 + row] = wkv;
    }
}

// ---------------------------------------------------------------------------
// Stage 3 (fused steps 4+5): out[n,c,t,v] = sum_k region[n,c,t,v,k]*wk[n,v,k] + b2
// One block per (n,c); wk for this n staged in LDS; each thread produces one
// (t,v) output from 5 aligned float2 loads (K=10 consecutive floats), so the
// block's loads tile the slab contiguously. Output stored nontemporally (never
// re-read) to preserve L2 for the 262MB region stream.
// ---------------------------------------------------------------------------
__global__ void vc_out_kernel(const float* __restrict__ region,
                              const float* __restrict__ wk2,  // (K, M)
                              const float* __restrict__ b2,   // (1)
                              float* __restrict__ out) {
    const int nc = blockIdx.x;       // n*C + c
    const int n  = nc / C_;

    __shared__ float wk_lds[V_ * K_];  // wk_lds[v*K + k]
    for (int i = threadIdx.x; i < V_ * K_; i += blockDim.x) {
        const int v = i / K_;
        const int k = i % K_;
        wk_lds[i] = wk2[(size_t)k * M_ + n * V_ + v];
    }
    __syncthreads();

    const float bias = b2[0];
    const float* slab = region + (size_t)nc * T_ * V_ * K_;
    float* oslab = out + (size_t)nc * T_ * V_;

    for (int i = threadIdx.x; i < T_ * V_; i += blockDim.x) {
        const int v = i % V_;
        const float2* p2 = (const float2*)(slab + (size_t)i * K_);  // 8B aligned
        const float* wv = &wk_lds[v * K_];
        float s = bias;
        #pragma unroll
        for (int j = 0; j < K_ / 2; ++j) {
            const float2 r = p2[j];
            s += r.x * wv[2 * j] + r.y * wv[2 * j + 1];
        }
        __builtin_nontemporal_store(s, &oslab[i]);
    }
}

// ---------------------------------------------------------------------------
extern "C" void kernel_launch(void* const* d_in, const int* in_sizes, int n_in,
                              void* d_out, int out_size, void* d_ws, size_t ws_size,
                              hipStream_t stream) {
    const float* region = (const float*)d_in[0];  // (N,C,T,V,K)
    const float* w1     = (const float*)d_in[1];  // (K,K,C)
    const float* b1     = (const float*)d_in[2];  // (K,K)
    const float* w2     = (const float*)d_in[3];  // (K)
    const float* b2     = (const float*)d_in[4];  // (1)
    float* out = (float*)d_out;                   // (N,C,T,V)

    // Workspace: mf2 (K*M*C = 512000 f32) then wk2 (K*M = 8000 f32) ~ 2.1 MB
    float* mf2 = (float*)d_ws;
    float* wk2 = mf2 + (size_t)K_ * M_ * C_;

    vc_mean_kernel<<<N_ * C_, 256, 0, stream>>>(region, mf2);
    vc_attn_kernel<<<K_ * (M_ / 16), 32, 0, stream>>>(mf2, w1, b1, w2, wk2);
    vc_out_kernel<<<N_ * C_, 256, 0, stream>>>(region, wk2, b2, out);
}